// GNNConvDropoutPool_16647293239654
// MI455X (gfx1250) — compile-verified
//
#include <hip/hip_runtime.h>

typedef __attribute__((ext_vector_type(2))) float v2f;
typedef __attribute__((ext_vector_type(8))) float v8f;

// ---------------- problem constants ----------------
constexpr int B   = 8;
constexpr int N0  = 16384;
constexpr int CIN = 64;
constexpr int D0  = 128;   // 2*HID
constexpr int D1  = 256;   // 4*HID
constexpr int EPG = N0 * 8;
constexpr int E   = B * EPG;        // 1,048,576 edges
constexpr int K1  = 13108;          // ceil(0.8*16384)
constexpr int K2  = 10487;          // ceil(0.8*13108)
constexpr int M1  = B * N0;         // 131072
constexpr int M2  = B * K1;         // 104864 (= 16*6554)

// ---------------- WMMA fp32 GEMM: C[M,N] = A[M,K] @ B[K,N], row-major ------
// One wave per 16x16 output tile; K stepped by 4 with V_WMMA_F32_16X16X4_F32.
__global__ void gemm_f32_wmma(const float* __restrict__ A,
                              const float* __restrict__ Bm,
                              float* __restrict__ C, int M, int K, int N) {
  const int lane = threadIdx.x;        // wave32
  const int mt   = blockIdx.x * 16;
  const int nt   = blockIdx.y * 16;
  const int half = lane >> 4;          // 0 or 1
  const int l16  = lane & 15;
  v8f acc = {};
  for (int k = 0; k < K; k += 4) {
    const int kb = k + half * 2;
    v2f a, b;
    // A 16x4 fragment: lane (half,l16) holds A[m][kb], A[m][kb+1]
    a.x = A[(long)(mt + l16) * K + kb];
    a.y = A[(long)(mt + l16) * K + kb + 1];
    // B 4x16 fragment: VGPR v holds B[2*half + v][n]
    b.x = Bm[(long)kb * N + nt + l16];
    b.y = Bm[(long)(kb + 1) * N + nt + l16];
    acc = __builtin_amdgcn_wmma_f32_16x16x4_f32(false, a, false, b,
                                                (short)0, acc, false, false);
  }
  for (int v = 0; v < 8; ++v) {
    const int row = mt + v + half * 8;
    C[(long)row * N + nt + l16] = acc[v];
  }
}

// ---------------- fills ----------------
__global__ void fill_f32(float* p, float v, long n) {
  long i = (long)blockIdx.x * blockDim.x + threadIdx.x;
  if (i < n) p[i] = v;
}
__global__ void fill_i32(int* p, int v, long n) {
  long i = (long)blockIdx.x * blockDim.x + threadIdx.x;
  if (i < n) p[i] = v;
}

// ---------------- GCN aggregation ----------------
__global__ void deg_accum(const int* __restrict__ dst,
                          const float* __restrict__ w,
                          float* __restrict__ deg, int e) {
  int i = blockIdx.x * blockDim.x + threadIdx.x;
  if (i >= e) return;
  float v = w ? w[i] : 1.0f;
  if (v != 0.0f) atomicAdd(&deg[dst[i]], v);
}

__global__ void norm_k(const int* __restrict__ src, const int* __restrict__ dst,
                       const float* __restrict__ w, const float* __restrict__ deg,
                       float* __restrict__ nrm, int e) {
  int i = blockIdx.x * blockDim.x + threadIdx.x;
  if (i >= e) return;
  float wv = w ? w[i] : 1.0f;
  nrm[i] = wv * rsqrtf(deg[src[i]]) * rsqrtf(deg[dst[i]]);
}

// one thread = one edge x 4 features; F/4 consecutive threads cover one edge row
__global__ void edge_agg(const int* __restrict__ src, const int* __restrict__ dst,
                         const float* __restrict__ nrm, const float* __restrict__ xw,
                         float* __restrict__ agg, int e, int F) {
  const int F4 = F >> 2;
  long gid = (long)blockIdx.x * blockDim.x + threadIdx.x;
  int ed = (int)(gid / F4);
  if (ed >= e) return;
  int c = ((int)(gid % F4)) << 2;
  float n = nrm[ed];
  if (n == 0.0f) return;
  const float4 v = *(const float4*)(xw + (long)src[ed] * F + c);
  float* o = agg + (long)dst[ed] * F + c;
  atomicAdd(o + 0, n * v.x);
  atomicAdd(o + 1, n * v.y);
  atomicAdd(o + 2, n * v.z);
  atomicAdd(o + 3, n * v.w);
}

// h = relu(agg + xw/deg + bias), in place over agg
__global__ void finalize_k(float* __restrict__ agg, const float* __restrict__ xw,
                           const float* __restrict__ deg, const float* __restrict__ bias,
                           long total, int F) {
  long i = (long)blockIdx.x * blockDim.x + threadIdx.x;
  if (i >= total) return;
  int node = (int)(i / F), f = (int)(i % F);
  float v = agg[i] + xw[i] / deg[node] + bias[f];
  agg[i] = v > 0.0f ? v : 0.0f;
}

// ---------------- TopK pooling ----------------
__global__ void invnorm_k(const float* __restrict__ p, int F, float* out) {
  __shared__ float red[256];
  float s = 0.0f;
  for (int i = threadIdx.x; i < F; i += 256) { float v = p[i]; s += v * v; }
  red[threadIdx.x] = s;
  __syncthreads();
  for (int o = 128; o > 0; o >>= 1) {
    if (threadIdx.x < o) red[threadIdx.x] += red[threadIdx.x + o];
    __syncthreads();
  }
  if (threadIdx.x == 0) out[0] = rsqrtf(red[0]);
}

__global__ void score_k(const float* __restrict__ h, const float* __restrict__ p,
                        const float* __restrict__ invn, float* __restrict__ sc,
                        int total, int F) {
  int i = blockIdx.x * blockDim.x + threadIdx.x;
  if (i >= total) return;
  const float* r = h + (long)i * F;
  float s = 0.0f;
  for (int f = 0; f < F; ++f) s += r[f] * p[f];
  sc[i] = s * invn[0];
}

__device__ __forceinline__ unsigned tokey(float f) {
  unsigned b = __float_as_uint(f);
  return (b & 0x80000000u) ? ~b : (b | 0x80000000u);  // ascending monotonic
}

// one block (256 thr) per graph; exact kth-largest via 4x8-bit MSB radix select
__global__ void radix_select(const float* __restrict__ score, int n, int k,
                             unsigned* __restrict__ thresh, int* __restrict__ ties) {
  __shared__ int hist[256];
  __shared__ unsigned prefix;
  __shared__ int remaining;
  const int g = blockIdx.x;
  const float* s = score + (long)g * n;
  if (threadIdx.x == 0) { prefix = 0u; remaining = k; }
  __syncthreads();
  for (int p = 3; p >= 0; --p) {
    hist[threadIdx.x] = 0;
    __syncthreads();
    const unsigned hm = (p == 3) ? 0u : (0xFFFFFFFFu << ((p + 1) * 8));
    for (int i = threadIdx.x; i < n; i += 256) {
      unsigned key = tokey(s[i]);
      if ((key & hm) == (prefix & hm))
        atomicAdd(&hist[(key >> (p * 8)) & 0xFF], 1);
    }
    __syncthreads();
    if (threadIdx.x == 0) {
      int rem = remaining;
      for (int b = 255; b >= 0; --b) {
        if (hist[b] >= rem) { prefix |= ((unsigned)b) << (p * 8); break; }
        rem -= hist[b];
      }
      remaining = rem;
    }
    __syncthreads();
  }
  if (threadIdx.x == 0) { thresh[g] = prefix; ties[g] = remaining; }
}

__global__ void compact_k(const float* __restrict__ h, const float* __restrict__ score,
                          const unsigned* __restrict__ thresh, const int* __restrict__ ties,
                          int* __restrict__ cnt, int* __restrict__ tcnt,
                          int* __restrict__ map, float* __restrict__ newx,
                          int n, int k, int F, int total) {
  int i = blockIdx.x * blockDim.x + threadIdx.x;
  if (i >= total) return;
  int g = i / n;
  unsigned key = tokey(score[i]);
  unsigned T = thresh[g];
  bool keep = false;
  if (key > T) keep = true;
  else if (key == T) keep = (atomicAdd(&tcnt[g], 1) < ties[g]);
  int m = -1;
  if (keep) {
    int pos = atomicAdd(&cnt[g], 1);
    m = g * k + pos;
    float sc = score[i];
    float r = sc > 0.0f ? sc : 0.0f;        // relu(score) gate
    const float* hr = h + (long)i * F;
    float* nr = newx + (long)m * F;
    for (int f = 0; f < F; ++f) nr[f] = hr[f] * r;
  }
  map[i] = m;
}

__global__ void remap_edges(const int* __restrict__ src, const int* __restrict__ dst,
                            const float* __restrict__ w, const int* __restrict__ map,
                            int* __restrict__ nsrc, int* __restrict__ ndst,
                            float* __restrict__ nw, int e) {
  int i = blockIdx.x * blockDim.x + threadIdx.x;
  if (i >= e) return;
  int a = map[src[i]], b = map[dst[i]];
  float wv = w ? w[i] : 1.0f;
  bool valid = (a >= 0) && (b >= 0) && (wv > 0.0f);
  nsrc[i] = valid ? a : 0;
  ndst[i] = valid ? b : 0;
  nw[i]   = valid ? 1.0f : 0.0f;
}

// ---------------- head ----------------
__global__ void meanpool_k(const float* __restrict__ h, float* __restrict__ g,
                           int k, int F) {
  int b = blockIdx.x, f = threadIdx.x;
  const float* base = h + (long)b * k * F + f;
  float s = 0.0f;
  for (int i = 0; i < k; ++i) s += base[(long)i * F];
  g[b * F + f] = s / (float)k;
}

__global__ void dense1_k(const float* __restrict__ g, const float* __restrict__ W,
                         const float* __restrict__ bias, float* __restrict__ out) {
  int j = threadIdx.x;  // 256
  for (int b = 0; b < B; ++b) {
    float s = bias[j];
    for (int i = 0; i < D1; ++i) s += g[b * D1 + i] * W[i * D1 + j];
    out[b * D1 + j] = s > 0.0f ? s : 0.0f;
  }
}

__global__ void final_k(const float* __restrict__ x1, const float* __restrict__ sex,
                        const float* __restrict__ cag, const float* __restrict__ W,
                        const float* __restrict__ bias, float* __restrict__ out) {
  __shared__ float sl[B][5];
  int t = threadIdx.x;
  if (t < B * 5) {
    int b = t / 5, c = t % 5;
    float s = bias[c];
    for (int i = 0; i < D1; ++i) s += x1[b * D1 + i] * W[i * 5 + c];
    s += sex[b] * W[D1 * 5 + c] + cag[b] * W[(D1 + 1) * 5 + c];
    sl[b][c] = s;
  }
  __syncthreads();
  if (t < B) {
    float mx = sl[t][0];
    for (int c = 1; c < 5; ++c) mx = fmaxf(mx, sl[t][c]);
    float e[5], sum = 0.0f;
    for (int c = 0; c < 5; ++c) { e[c] = __expf(sl[t][c] - mx); sum += e[c]; }
    for (int c = 0; c < 5; ++c) out[t * 5 + c] = e[c] / sum;
  }
}

// ---------------- launcher ----------------
extern "C" void kernel_launch(void* const* d_in, const int* in_sizes, int n_in,
                              void* d_out, int out_size, void* d_ws, size_t ws_size,
                              hipStream_t stream) {
  (void)in_sizes; (void)n_in; (void)out_size; (void)ws_size;
  const float* x   = (const float*)d_in[0];
  const float* sex = (const float*)d_in[1];
  const float* cag = (const float*)d_in[2];
  const float* Wc0 = (const float*)d_in[3];
  const float* bc0 = (const float*)d_in[4];
  const float* pw0 = (const float*)d_in[5];
  const float* Wc1 = (const float*)d_in[6];
  const float* bc1 = (const float*)d_in[7];
  const float* pw1 = (const float*)d_in[8];
  const float* lw1 = (const float*)d_in[9];
  const float* lb1 = (const float*)d_in[10];
  const float* lw2 = (const float*)d_in[11];
  const float* lb2 = (const float*)d_in[12];
  const int*  esrc = (const int*)d_in[13];
  const int*  edst = (const int*)d_in[14];
  float* out = (float*)d_out;

  char* ws = (char*)d_ws;
  size_t off = 0;
  auto alloc = [&](size_t bytes) -> char* {
    char* p = ws + off;
    off += (bytes + 255) & ~(size_t)255;
    return p;
  };
  float* xw0  = (float*)alloc((size_t)M1 * D0 * 4);  // 64MB; reused as h1 (pooled)
  float* agg0 = (float*)alloc((size_t)M1 * D0 * 4);  // 64MB; becomes h0 in place
  float* big1 = (float*)alloc((size_t)M2 * D1 * 4);  // 102MB; xw1, later h2
  float* big2 = (float*)alloc((size_t)M2 * D1 * 4);  // 102MB; agg1 -> h1b in place
  float* deg  = (float*)alloc((size_t)M1 * 4);
  float* nrm  = (float*)alloc((size_t)E * 4);
  float* sco  = (float*)alloc((size_t)M1 * 4);
  int*   src1 = (int*)  alloc((size_t)E * 4);
  int*   dst1 = (int*)  alloc((size_t)E * 4);
  float* w1   = (float*)alloc((size_t)E * 4);
  int*   map  = (int*)  alloc((size_t)M1 * 4);
  char*  sm   = alloc(4096);
  float*    invn   = (float*)sm;
  unsigned* thresh = (unsigned*)(sm + 64);
  int*      ties   = (int*)(sm + 128);
  int*      cnt    = (int*)(sm + 192);
  int*      tcnt   = (int*)(sm + 256);
  float* gbuf = (float*)alloc(B * D1 * 4);
  float* x1b  = (float*)alloc(B * D1 * 4);

  auto nb = [](long n, int t) { return (unsigned)((n + t - 1) / t); };

  // -------- conv1: h0 = relu(GCN(x)) --------
  gemm_f32_wmma<<<dim3(M1 / 16, D0 / 16), 32, 0, stream>>>(x, Wc0, xw0, M1, CIN, D0);
  fill_f32<<<nb(M1, 256), 256, 0, stream>>>(deg, 1.0f, M1);
  fill_f32<<<nb((long)M1 * D0, 256), 256, 0, stream>>>(agg0, 0.0f, (long)M1 * D0);
  deg_accum<<<nb(E, 256), 256, 0, stream>>>(edst, nullptr, deg, E);
  norm_k<<<nb(E, 256), 256, 0, stream>>>(esrc, edst, nullptr, deg, nrm, E);
  edge_agg<<<nb((long)E * (D0 / 4), 256), 256, 0, stream>>>(esrc, edst, nrm, xw0, agg0, E, D0);
  finalize_k<<<nb((long)M1 * D0, 256), 256, 0, stream>>>(agg0, xw0, deg, bc0, (long)M1 * D0, D0);

  // -------- pool1 --------
  invnorm_k<<<1, 256, 0, stream>>>(pw0, D0, invn);
  score_k<<<nb(M1, 256), 256, 0, stream>>>(agg0, pw0, invn, sco, M1, D0);
  radix_select<<<B, 256, 0, stream>>>(sco, N0, K1, thresh, ties);
  fill_i32<<<nb(M1, 256), 256, 0, stream>>>(map, -1, M1);
  fill_i32<<<1, 32, 0, stream>>>(cnt, 0, B);
  fill_i32<<<1, 32, 0, stream>>>(tcnt, 0, B);
  compact_k<<<nb(M1, 256), 256, 0, stream>>>(agg0, sco, thresh, ties, cnt, tcnt, map,
                                             xw0 /*h1*/, N0, K1, D0, M1);
  remap_edges<<<nb(E, 256), 256, 0, stream>>>(esrc, edst, nullptr, map, src1, dst1, w1, E);

  // -------- conv2: h1b = relu(GCN(h1)) --------
  gemm_f32_wmma<<<dim3(M2 / 16, D1 / 16), 32, 0, stream>>>(xw0, Wc1, big1, M2, D0, D1);
  fill_f32<<<nb(M2, 256), 256, 0, stream>>>(deg, 1.0f, M2);
  fill_f32<<<nb((long)M2 * D1, 256), 256, 0, stream>>>(big2, 0.0f, (long)M2 * D1);
  deg_accum<<<nb(E, 256), 256, 0, stream>>>(dst1, w1, deg, E);
  norm_k<<<nb(E, 256), 256, 0, stream>>>(src1, dst1, w1, deg, nrm, E);
  edge_agg<<<nb((long)E * (D1 / 4), 256), 256, 0, stream>>>(src1, dst1, nrm, big1, big2, E, D1);
  finalize_k<<<nb((long)M2 * D1, 256), 256, 0, stream>>>(big2, big1, deg, bc1, (long)M2 * D1, D1);

  // -------- pool2 (post-pool edges are unused downstream; skip remap) --------
  invnorm_k<<<1, 256, 0, stream>>>(pw1, D1, invn);
  score_k<<<nb(M2, 256), 256, 0, stream>>>(big2, pw1, invn, sco, M2, D1);
  radix_select<<<B, 256, 0, stream>>>(sco, K1, K2, thresh, ties);
  fill_i32<<<nb(M2, 256), 256, 0, stream>>>(map, -1, M2);
  fill_i32<<<1, 32, 0, stream>>>(cnt, 0, B);
  fill_i32<<<1, 32, 0, stream>>>(tcnt, 0, B);
  compact_k<<<nb(M2, 256), 256, 0, stream>>>(big2, sco, thresh, ties, cnt, tcnt, map,
                                             big1 /*h2*/, K1, K2, D1, M2);

  // -------- head --------
  meanpool_k<<<B, D1, 0, stream>>>(big1, gbuf, K2, D1);
  dense1_k<<<1, D1, 0, stream>>>(gbuf, lw1, lb1, x1b);
  final_k<<<1, 64, 0, stream>>>(x1b, sex, cag, lw2, lb2, out);
}